// GCNII_80178449482260
// MI455X (gfx1250) — compile-verified
//
#include <hip/hip_runtime.h>
#include <math.h>

typedef __attribute__((ext_vector_type(2))) float v2f;
typedef __attribute__((ext_vector_type(8))) float v8f;

#define KC 16            // K-chunk staged in LDS per iteration
#define BROW 80          // padded LDS row (floats): disjoint bank ranges for the
                         // two half-wave row reads (16*k mod 64 vs 16*k+32 mod 64)

// ---------------------------------------------------------------------------
// fp32 WMMA GEMM:  D = act( sA*(A@B) + sB*E1 + sC*E2 + bias )
//   A: [M,K] lda,  B: [K,N] ldb,  D: [M,N] ldd (column-offset writes allowed),
//   E1/E2: [M,N] stride N, bias: [N].  act: 0=identity, 1=leaky, 2=elu.
//
// Block = 4 waves stacked in M: block tile 128(M) x 64(N), wave tile 32x64.
// B tile [KC x 64] is shared by all 4 waves and double-buffered in LDS via
// CDNA5 async global->LDS copies (ASYNCcnt + s_wait_asynccnt + barrier).
// Per k-step each wave issues 2 A b64 loads + 8 LDS reads + 8 WMMAs.
// Grid sized exactly; all WMMA-adjacent branches are wave-uniform (EXEC ones).
// ---------------------------------------------------------------------------
__global__ __launch_bounds__(128)
void gemm128x64_f32(const float* __restrict__ A, const float* __restrict__ B,
                    float* __restrict__ D,
                    int N, int K, int lda, int ldb, int ldd,
                    const float* __restrict__ E1, const float* __restrict__ E2,
                    const float* __restrict__ bias,
                    float sA, float sB, float sC, int act, float slope,
                    int nTiles /* = N/64 */) {
  __shared__ __align__(16) float Bsm[2][KC][BROW];

  const int wave  = threadIdx.x >> 5;
  const int lane  = threadIdx.x & 31;
  const int half  = lane >> 4;          // K-half of the fragment this lane holds
  const int l15   = lane & 15;
  const int tile  = blockIdx.x;
  const int mBase = (tile / nTiles) * 128 + wave * 32;
  const int nBase = (tile % nTiles) * 64;

  // --- async B staging: 16x64 floats = 256 b128 transfers, 2 per thread ----
  const int tid = threadIdx.x;
  const int sr0 = tid >> 4;                  // rows 0..7
  const int sc  = (tid & 15) * 4;            // col group
  const int sr1 = sr0 + 8;                   // rows 8..15
  const float* __restrict__ Bst = B + nBase + sc;

  auto stage = [&](int k0, int buf) {
    const float* g0 = Bst + (size_t)(k0 + sr0) * ldb;
    const float* g1 = Bst + (size_t)(k0 + sr1) * ldb;
    unsigned l0 = (unsigned)(uintptr_t)&Bsm[buf][sr0][sc];
    unsigned l1 = (unsigned)(uintptr_t)&Bsm[buf][sr1][sc];
    asm volatile("global_load_async_to_lds_b128 %0, %1, off"
                 :: "v"(l0), "v"(g0) : "memory");
    asm volatile("global_load_async_to_lds_b128 %0, %1, off"
                 :: "v"(l1), "v"(g1) : "memory");
  };

  // A fragment row pointers (per-lane rows; served by L2, no cross-wave reuse)
  const float* __restrict__ arow0 = A + (size_t)(mBase + l15) * lda + 2 * half;
  const float* __restrict__ arow1 = arow0 + (size_t)16 * lda;

  v8f a00 = {}, a01 = {}, a02 = {}, a03 = {};   // M-subtile 0, N-subtiles 0..3
  v8f a10 = {}, a11 = {}, a12 = {}, a13 = {};   // M-subtile 1

  const int nchunk = K / KC;
  stage(0, 0);
  asm volatile("s_wait_asynccnt 0x0" ::: "memory");
  __syncthreads();

  int p = 0;
  for (int c = 0; c < nchunk; ++c) {
    if (c + 1 < nchunk) stage((c + 1) * KC, p ^ 1);   // overlap with compute

    const int k0 = c * KC;
    #pragma unroll
    for (int kk = 0; kk < KC; kk += 4) {
      v2f fa0 = *(const v2f*)(arow0 + k0 + kk);
      v2f fa1 = *(const v2f*)(arow1 + k0 + kk);
      const int kb = kk + 2 * half;
      v2f b0, b1, b2, b3;
      b0.x = Bsm[p][kb][l15 +  0]; b0.y = Bsm[p][kb + 1][l15 +  0];
      b1.x = Bsm[p][kb][l15 + 16]; b1.y = Bsm[p][kb + 1][l15 + 16];
      b2.x = Bsm[p][kb][l15 + 32]; b2.y = Bsm[p][kb + 1][l15 + 32];
      b3.x = Bsm[p][kb][l15 + 48]; b3.y = Bsm[p][kb + 1][l15 + 48];
      a00 = __builtin_amdgcn_wmma_f32_16x16x4_f32(false, fa0, false, b0, (short)0, a00, false, false);
      a01 = __builtin_amdgcn_wmma_f32_16x16x4_f32(false, fa0, false, b1, (short)0, a01, false, false);
      a02 = __builtin_amdgcn_wmma_f32_16x16x4_f32(false, fa0, false, b2, (short)0, a02, false, false);
      a03 = __builtin_amdgcn_wmma_f32_16x16x4_f32(false, fa0, false, b3, (short)0, a03, false, false);
      a10 = __builtin_amdgcn_wmma_f32_16x16x4_f32(false, fa1, false, b0, (short)0, a10, false, false);
      a11 = __builtin_amdgcn_wmma_f32_16x16x4_f32(false, fa1, false, b1, (short)0, a11, false, false);
      a12 = __builtin_amdgcn_wmma_f32_16x16x4_f32(false, fa1, false, b2, (short)0, a12, false, false);
      a13 = __builtin_amdgcn_wmma_f32_16x16x4_f32(false, fa1, false, b3, (short)0, a13, false, false);
    }

    // my async copies for chunk c+1 done; barrier publishes them to all waves
    // and guarantees everyone finished reading buffer p before its reuse.
    asm volatile("s_wait_asynccnt 0x0" ::: "memory");
    __syncthreads();
    p ^= 1;
  }

  // Epilogue. C/D layout: acc[r] -> (M = mBase+16*mi + r + 8*half, N = nBase+16*t+l15)
  auto store_tile = [&](v8f acc, int mi, int t) {
    const int n = nBase + 16 * t + l15;
    #pragma unroll
    for (int r = 0; r < 8; ++r) {
      const int m = mBase + 16 * mi + r + 8 * half;
      float v = sA * acc[r];
      if (E1)   v += sB * E1[(size_t)m * N + n];
      if (E2)   v += sC * E2[(size_t)m * N + n];
      if (bias) v += bias[n];
      if (act == 1)      v = (v >= 0.0f) ? v : slope * v;   // LeakyReLU
      else if (act == 2) v = (v > 0.0f)  ? v : expm1f(v);   // ELU
      D[(size_t)m * ldd + n] = v;
    }
  };
  store_tile(a00, 0, 0); store_tile(a01, 0, 1); store_tile(a02, 0, 2); store_tile(a03, 0, 3);
  store_tile(a10, 1, 0); store_tile(a11, 1, 1); store_tile(a12, 1, 2); store_tile(a13, 1, 3);
}

// ---------------------------------------------------------------------------
// Attention score vectors: s1[i] = Wh[i,:] . a[0:Fo],  s2[i] = Wh[i,:] . a[Fo:2Fo]
// ---------------------------------------------------------------------------
__global__ __launch_bounds__(256)
void rowdot2_kernel(const float* __restrict__ Wh, const float* __restrict__ a,
                    int Fo, float* __restrict__ s1, float* __restrict__ s2) {
  __shared__ float sm1[256];
  __shared__ float sm2[256];
  const int row = blockIdx.x;
  const int t = threadIdx.x;
  const float* __restrict__ w = Wh + (size_t)row * Fo;
  float p1 = 0.0f, p2 = 0.0f;
  for (int j = t; j < Fo; j += 256) {
    const float wv = w[j];
    p1 += wv * a[j];
    p2 += wv * a[Fo + j];
  }
  sm1[t] = p1; sm2[t] = p2; __syncthreads();
  for (int s = 128; s > 0; s >>= 1) {
    if (t < s) { sm1[t] += sm1[t + s]; sm2[t] += sm2[t + s]; }
    __syncthreads();
  }
  if (t == 0) { s1[row] = sm1[0]; s2[row] = sm2[0]; }
}

// ---------------------------------------------------------------------------
// att[i,:] = softmax_j( mask ? leaky(s1[i]+s2[j], 0.01) : -9e15 )
// ---------------------------------------------------------------------------
__global__ __launch_bounds__(256)
void masked_softmax_kernel(const float* __restrict__ adj,
                           const float* __restrict__ s1,
                           const float* __restrict__ s2,
                           float* __restrict__ att, int Ncols) {
  __shared__ float sm[256];
  const int row = blockIdx.x;
  const int t = threadIdx.x;
  const float* __restrict__ arow = adj + (size_t)row * Ncols;
  float* __restrict__ orow = att + (size_t)row * Ncols;
  const float si = s1[row];

  float mx = -3.0e38f;
  for (int j = t; j < Ncols; j += 256) {
    float e;
    if (arow[j] > 0.0f) { e = si + s2[j]; e = (e >= 0.0f) ? e : 0.01f * e; }
    else                { e = -9.0e15f; }
    mx = fmaxf(mx, e);
  }
  sm[t] = mx; __syncthreads();
  for (int s = 128; s > 0; s >>= 1) { if (t < s) sm[t] = fmaxf(sm[t], sm[t + s]); __syncthreads(); }
  mx = sm[0]; __syncthreads();

  float sum = 0.0f;
  for (int j = t; j < Ncols; j += 256) {
    float p = 0.0f;
    if (arow[j] > 0.0f) {
      float e = si + s2[j]; e = (e >= 0.0f) ? e : 0.01f * e;
      p = expf(e - mx);
    }
    orow[j] = p;
    sum += p;
  }
  sm[t] = sum; __syncthreads();
  for (int s = 128; s > 0; s >>= 1) { if (t < s) sm[t] += sm[t + s]; __syncthreads(); }
  const float inv = 1.0f / sm[0];
  for (int j = t; j < Ncols; j += 256) orow[j] *= inv;
}

// ---------------------------------------------------------------------------
static inline void launch_gemm(hipStream_t st, const float* A, const float* B, float* D,
                               int M, int N, int K, int lda, int ldb, int ldd,
                               const float* E1, const float* E2, const float* bias,
                               float sA, float sB, float sC, int act, float slope) {
  const int nTiles = N / 64;
  const int blocks = (M / 128) * nTiles;     // M is always 1024 here
  gemm128x64_f32<<<dim3(blocks), dim3(128), 0, st>>>(
      A, B, D, N, K, lda, ldb, ldd, E1, E2, bias, sA, sB, sC, act, slope, nTiles);
}

extern "C" void kernel_launch(void* const* d_in, const int* in_sizes, int n_in,
                              void* d_out, int out_size, void* d_ws, size_t ws_size,
                              hipStream_t stream) {
  (void)in_sizes; (void)n_in; (void)out_size; (void)ws_size;
  const float* x    = (const float*)d_in[0];
  const float* adj  = (const float*)d_in[1];
  const float* Wg1  = (const float*)d_in[2];   // [5,1024,1024]
  const float* ag1  = (const float*)d_in[3];   // [5,2048,1]
  const float* Wo1  = (const float*)d_in[4];   // [5120,512]
  const float* ao1  = (const float*)d_in[5];   // [1024,1]
  const float* Wg2  = (const float*)d_in[6];   // [2,512,512]
  const float* ag2  = (const float*)d_in[7];   // [2,1024,1]
  const float* Wo2  = (const float*)d_in[8];   // [1024,1024]
  const float* ao2  = (const float*)d_in[9];   // [2048,1]
  const float* fc0w = (const float*)d_in[10];  // [1024,512]
  const float* fc0b = (const float*)d_in[11];  // [512]
  const float* fc1w = (const float*)d_in[12];  // [512,1024]
  const float* fc1b = (const float*)d_in[13];  // [1024]
  // d_in[14] (cw0) is dead in the reference: only the last conv layer survives.
  const float* cw1  = (const float*)d_in[15];  // [1024,1024]
  float* out = (float*)d_out;

  // Workspace layout (floats): ~38 MB total
  float* WS   = (float*)d_ws;
  float* WH   = WS;                                   // 1024*1024
  float* ATT  = WS + (1u << 20);                      // 1024*1024
  float* HCAT = WS + (2u << 20);                      // 1024*5120 (reused for hcat2)
  float* XG   = HCAT + (size_t)1024 * 5120;           // 1024*1024 (xg1 then xg2)
  float* H0   = XG + (1u << 20);                      // 1024*512
  float* SUP  = H0 + (size_t)1024 * 512;              // 1024*512
  float* S1   = SUP + (size_t)1024 * 512;             // 1024
  float* S2   = S1 + 1024;                            // 1024

  // ===== GAT layer 1: 5 heads, concat -> [1024, 5120] ======================
  for (int h = 0; h < 5; ++h) {
    launch_gemm(stream, x, Wg1 + (size_t)h * 1024 * 1024, WH,
                1024, 1024, 1024, 1024, 1024, 1024,
                nullptr, nullptr, nullptr, 1.0f, 0.0f, 0.0f, 0, 0.0f);
    rowdot2_kernel<<<1024, 256, 0, stream>>>(WH, ag1 + (size_t)h * 2048, 1024, S1, S2);
    masked_softmax_kernel<<<1024, 256, 0, stream>>>(adj, S1, S2, ATT, 1024);
    launch_gemm(stream, ATT, WH, HCAT + (size_t)h * 1024,
                1024, 1024, 1024, 1024, 1024, 5120,
                nullptr, nullptr, nullptr, 1.0f, 0.0f, 0.0f, 2, 0.0f);
  }
  // out attention: xg1 = elu(att @ (hcat @ Wo1))   [1024,512]
  launch_gemm(stream, HCAT, Wo1, WH, 1024, 512, 5120, 5120, 512, 512,
              nullptr, nullptr, nullptr, 1.0f, 0.0f, 0.0f, 0, 0.0f);
  rowdot2_kernel<<<1024, 256, 0, stream>>>(WH, ao1, 512, S1, S2);
  masked_softmax_kernel<<<1024, 256, 0, stream>>>(adj, S1, S2, ATT, 1024);
  launch_gemm(stream, ATT, WH, XG, 1024, 512, 1024, 1024, 512, 512,
              nullptr, nullptr, nullptr, 1.0f, 0.0f, 0.0f, 2, 0.0f);

  // ===== GAT layer 2: 2 heads, concat -> [1024, 1024] ======================
  for (int h = 0; h < 2; ++h) {
    launch_gemm(stream, XG, Wg2 + (size_t)h * 512 * 512, WH,
                1024, 512, 512, 512, 512, 512,
                nullptr, nullptr, nullptr, 1.0f, 0.0f, 0.0f, 0, 0.0f);
    rowdot2_kernel<<<1024, 256, 0, stream>>>(WH, ag2 + (size_t)h * 1024, 512, S1, S2);
    masked_softmax_kernel<<<1024, 256, 0, stream>>>(adj, S1, S2, ATT, 1024);
    launch_gemm(stream, ATT, WH, HCAT + (size_t)h * 512,
                1024, 512, 1024, 1024, 512, 1024,
                nullptr, nullptr, nullptr, 1.0f, 0.0f, 0.0f, 2, 0.0f);
  }
  // out attention: xg2 = elu(att @ (hcat2 @ Wo2))  [1024,1024]
  launch_gemm(stream, HCAT, Wo2, WH, 1024, 1024, 1024, 1024, 1024, 1024,
              nullptr, nullptr, nullptr, 1.0f, 0.0f, 0.0f, 0, 0.0f);
  rowdot2_kernel<<<1024, 256, 0, stream>>>(WH, ao2, 1024, S1, S2);
  masked_softmax_kernel<<<1024, 256, 0, stream>>>(adj, S1, S2, ATT, 1024);
  launch_gemm(stream, ATT, WH, XG, 1024, 1024, 1024, 1024, 1024, 1024,
              nullptr, nullptr, nullptr, 1.0f, 0.0f, 0.0f, 2, 0.0f);

  // ===== GCNII branch ======================================================
  // h0 = leaky(x @ fc0_w + fc0_b, 0.25)                      [1024,512]
  launch_gemm(stream, x, fc0w, H0, 1024, 512, 1024, 1024, 512, 512,
              nullptr, nullptr, fc0b, 1.0f, 0.0f, 0.0f, 1, 0.25f);
  // support = 0.9*(xg2 @ h0) + 0.1*h0                        [1024,512]
  launch_gemm(stream, XG, H0, SUP, 1024, 512, 1024, 1024, 512, 512,
              H0, nullptr, nullptr, 0.9f, 0.1f, 0.0f, 0, 0.0f);
  // h = leaky(theta*(cw1 @ support) + (1-theta)*support + h0, 0.25)
  const float theta = 0.22314355f;  // log(LAMDA/2 + 1) = log(1.25)
  launch_gemm(stream, cw1, SUP, WH, 1024, 512, 1024, 1024, 512, 512,
              SUP, H0, nullptr, theta, 1.0f - theta, 1.0f, 1, 0.25f);
  // out = h @ fc1_w + fc1_b                                  [1024,1024]
  launch_gemm(stream, WH, fc1w, out, 1024, 1024, 512, 512, 1024, 1024,
              nullptr, nullptr, fc1b, 1.0f, 0.0f, 0.0f, 0, 0.0f);
}